// LIF_49563922596596
// MI455X (gfx1250) — compile-verified
//
#include <hip/hip_runtime.h>
#include <cstdint>

// LIF scan over T with async global->LDS prefetch pipeline (gfx1250 ASYNCcnt path).
// I: [B=128, T=64, D=4096] f32 -> S same shape. Memory-bound: 268 MB, ~11.5 us @ 23.3 TB/s.

typedef __attribute__((ext_vector_type(4))) float v4f;

#define LIF_T   64
#define LIF_D   4096
#define STAGES  8          // per-wave async FIFO depth: 8 x 512B in flight per wave
#define BETA    0.9f
#define THETA   1.0f

__global__ __launch_bounds__(256)
void lif_async_kernel(const float* __restrict__ I, float* __restrict__ S)
{
    constexpr int D4 = LIF_D / 4;              // 1024 float4 positions per (b, t)
    // [wave][stage][lane] : 8 waves * 8 stages * 32 lanes * 16B = 32 KB LDS / block
    __shared__ v4f smem[8][STAGES][32];

    const int lane = threadIdx.x & 31;
    const int wave = threadIdx.x >> 5;
    const int g    = blockIdx.x * 256 + threadIdx.x;   // 0 .. B*D4-1
    const int d4   = g % D4;
    const int b    = g / D4;

    const float* gI = I + (size_t)b * LIF_T * LIF_D + (size_t)d4 * 4;
    float*       gS = S + (size_t)b * LIF_T * LIF_D + (size_t)d4 * 4;

    // Per-lane LDS destination byte-addresses for each FIFO stage.
    unsigned lds[STAGES];
#pragma unroll
    for (int s = 0; s < STAGES; ++s)
        lds[s] = (unsigned)(uintptr_t)&smem[wave][s][lane];

    // Prologue: fill the async FIFO with timesteps 0..STAGES-1 (ASYNCcnt = STAGES).
#pragma unroll
    for (int t = 0; t < STAGES; ++t) {
        uint64_t ga = (uint64_t)(uintptr_t)(gI + (size_t)t * LIF_D);
        asm volatile("global_load_async_to_lds_b128 %0, %1, off th:TH_LOAD_NT"
                     :: "v"(lds[t]), "v"(ga) : "memory");
    }

    v4f U = {0.0f, 0.0f, 0.0f, 0.0f};

    // Steady state: consume stage t%8, refill it with timestep t+8.
#pragma unroll 8
    for (int t = 0; t < LIF_T - STAGES; ++t) {
        // Oldest of the 8 outstanding async loads (timestep t) is complete when cnt<=7.
        asm volatile("s_wait_asynccnt 0x7" ::: "memory");
        v4f It = smem[wave][t % STAGES][lane];
        // Ensure the ds_load retired before the async DMA may overwrite this stage.
        asm volatile("s_wait_dscnt 0x0" ::: "memory");
        uint64_t ga = (uint64_t)(uintptr_t)(gI + (size_t)(t + STAGES) * LIF_D);
        asm volatile("global_load_async_to_lds_b128 %0, %1, off th:TH_LOAD_NT"
                     :: "v"(lds[t % STAGES]), "v"(ga) : "memory");

        // LIF update (fp32, exact threshold semantics).
        U = BETA * U + (1.0f - BETA) * It;
        v4f Sv;
        Sv.x = (U.x >= THETA) ? 1.0f : 0.0f;
        Sv.y = (U.y >= THETA) ? 1.0f : 0.0f;
        Sv.z = (U.z >= THETA) ? 1.0f : 0.0f;
        Sv.w = (U.w >= THETA) ? 1.0f : 0.0f;
        U = U - Sv * THETA;

        __builtin_nontemporal_store(Sv, (v4f*)(gS + (size_t)t * LIF_D));
    }

    // Drain: everything for t = T-8 .. T-1 is already (or will be) resident after cnt==0.
    asm volatile("s_wait_asynccnt 0x0" ::: "memory");
#pragma unroll
    for (int t = LIF_T - STAGES; t < LIF_T; ++t) {
        v4f It = smem[wave][t % STAGES][lane];
        U = BETA * U + (1.0f - BETA) * It;
        v4f Sv;
        Sv.x = (U.x >= THETA) ? 1.0f : 0.0f;
        Sv.y = (U.y >= THETA) ? 1.0f : 0.0f;
        Sv.z = (U.z >= THETA) ? 1.0f : 0.0f;
        Sv.w = (U.w >= THETA) ? 1.0f : 0.0f;
        U = U - Sv * THETA;
        __builtin_nontemporal_store(Sv, (v4f*)(gS + (size_t)t * LIF_D));
    }
}

extern "C" void kernel_launch(void* const* d_in, const int* in_sizes, int n_in,
                              void* d_out, int out_size, void* d_ws, size_t ws_size,
                              hipStream_t stream)
{
    (void)n_in; (void)out_size; (void)d_ws; (void)ws_size;
    const float* I = (const float*)d_in[0];
    float*       S = (float*)d_out;

    const int total    = in_sizes[0];            // B*T*D = 128*64*4096
    const int nthreads = total / LIF_T / 4;      // B * D/4 = 131072
    const int nblocks  = nthreads / 256;         // 512 blocks of 8 wave32s

    lif_async_kernel<<<nblocks, 256, 0, stream>>>(I, S);
}